// Net_54838142435975
// MI455X (gfx1250) — compile-verified
//
#include <hip/hip_runtime.h>
#include <hip/hip_bf16.h>
#include <cstddef>

#define HH 128
#define WW 128
#define HW (HH*WW)
#define LTOK HW
#define CF 32
#define DI 64
#define DSTATE 8
#define NBLK(n) (((n)+255)/256)

typedef float v2f __attribute__((ext_vector_type(2)));
typedef float v8f __attribute__((ext_vector_type(8)));

__device__ __forceinline__ float siluf(float v){ return v / (1.f + expf(-v)); }

// ---------------------------------------------------------------- WMMA GEMM
// C[m,n] = sum_k A[m*sAm + k*sAk] * B[k*sBk + n*sBn] (+ bias[n])
// One wave computes a 16 x (16*NT) strip: A fragment loaded once per K-step
// and reused across NT accumulators (4x less A traffic, 4x WMMA density).
// fp32 WMMA 16x16x4. M%16==0, N%(16*NT)==0, K%4==0.
template<int NT>
__global__ void k_wmma_gemm_nt(const float* __restrict__ A, int sAm, int sAk,
                               const float* __restrict__ B, int sBk, int sBn,
                               const float* __restrict__ bias,
                               float* __restrict__ C, int sCm, int sCn, int K)
{
    const int lane = threadIdx.x;          // 0..31
    const int m0 = blockIdx.x * 16;
    const int n0 = blockIdx.y * (16 * NT);
    const int half = lane >> 4;            // 0 or 1
    const int l16  = lane & 15;
    const int rowA = m0 + l16;
    v8f acc[NT];
#pragma unroll
    for (int t = 0; t < NT; ++t) acc[t] = 0.f;
    for (int kk = 0; kk < K; kk += 4) {
        const int kb = kk + half * 2;
        v2f a;
        a[0] = A[(size_t)rowA * sAm + (size_t)(kb + 0) * sAk];
        a[1] = A[(size_t)rowA * sAm + (size_t)(kb + 1) * sAk];
#pragma unroll
        for (int t = 0; t < NT; ++t) {
            const int colB = n0 + t * 16 + l16;
            v2f b;
            b[0] = B[(size_t)(kb + 0) * sBk + (size_t)colB * sBn];
            b[1] = B[(size_t)(kb + 1) * sBk + (size_t)colB * sBn];
            acc[t] = __builtin_amdgcn_wmma_f32_16x16x4_f32(false, a, false, b,
                                                           (short)0, acc[t], false, false);
        }
    }
#pragma unroll
    for (int t = 0; t < NT; ++t) {
        const int colB = n0 + t * 16 + l16;
        const float bv = bias ? bias[colB] : 0.f;
#pragma unroll
        for (int v = 0; v < 8; ++v) {
            const int m = m0 + v + 8 * half;
            C[(size_t)m * sCm + (size_t)colB * sCn] = acc[t][v] + bv;
        }
    }
}

// ---------------------------------------------------------------- elementwise
__global__ void k_zero(float* o, int n){ int i=blockIdx.x*256+threadIdx.x; if(i<n) o[i]=0.f; }
__global__ void k_add(const float* a, const float* b, float* o, int n){
    int i=blockIdx.x*256+threadIdx.x; if(i<n) o[i]=a[i]+b[i];
}
__global__ void k_upsample4(const float* ms, float* o){
    int i=blockIdx.x*256+threadIdx.x; if(i>=4*HW) return;
    int c=i/HW, p=i-c*HW, y=p/WW, x=p-y*WW;
    o[i] = ms[c*(32*32) + (y>>2)*32 + (x>>2)];
}
__global__ void k_tok_to_img(const float* t, float* img){
    int i=blockIdx.x*256+threadIdx.x; if(i>=CF*HW) return;
    int c=i/HW, p=i-c*HW; img[i]=t[p*CF+c];
}
__global__ void k_img_to_tok(const float* img, float* t){
    int i=blockIdx.x*256+threadIdx.x; if(i>=LTOK*CF) return;
    int tt=i/CF, c=i-tt*CF; t[i]=img[c*HW+tt];
}
__global__ void k_swap_mix(const float* msn, const float* pann, float* ms_s, float* pan_s){
    int i=blockIdx.x*256+threadIdx.x; if(i>=LTOK*CF) return;
    int c=i%CF;
    if (c < CF/2){ ms_s[i]=pann[i]; pan_s[i]=msn[i]; }
    else         { ms_s[i]=msn[i];  pan_s[i]=pann[i]; }
}
// ---------------------------------------------------------------- convs
__global__ void k_conv3x3(const float* __restrict__ in, const float* __restrict__ w,
                          const float* __restrict__ bias, const float* __restrict__ resid,
                          float* __restrict__ out, int Cin, int Cout, int act)
{
    int idx=blockIdx.x*256+threadIdx.x; if(idx>=Cout*HW) return;
    int co=idx/HW, p=idx-co*HW, y=p/WW, x=p-y*WW;
    float acc = bias[co];
    const float* wp = w + (size_t)co*Cin*9;
    for (int ci=0; ci<Cin; ++ci){
        const float* ip = in + (size_t)ci*HW;
#pragma unroll
        for (int ky=0; ky<3; ++ky){
            int iy=y+ky-1; if((unsigned)iy>=HH) continue;
#pragma unroll
            for (int kx=0; kx<3; ++kx){
                int ix=x+kx-1; if((unsigned)ix>=WW) continue;
                acc += ip[iy*WW+ix]*wp[ci*9+ky*3+kx];
            }
        }
    }
    if (act==1) acc = acc>=0.f ? acc : 0.2f*acc;
    else if (act==2) acc = fmaxf(acc,0.f);
    if (resid) acc += resid[idx];
    out[idx]=acc;
}
__global__ void k_dwconv_add(const float* __restrict__ in, const float* __restrict__ w,
                             const float* __restrict__ bias, float* __restrict__ out)
{
    int idx=blockIdx.x*256+threadIdx.x; if(idx>=CF*HW) return;
    int co=idx/HW, p=idx-co*HW, y=p/WW, x=p-y*WW;
    float acc = bias[co];
    const float* ip = in + (size_t)co*HW;
#pragma unroll
    for (int ky=0; ky<3; ++ky){
        int iy=y+ky-1; if((unsigned)iy>=HH) continue;
#pragma unroll
        for (int kx=0; kx<3; ++kx){
            int ix=x+kx-1; if((unsigned)ix>=WW) continue;
            acc += ip[iy*WW+ix]*w[co*9+ky*3+kx];
        }
    }
    out[idx] = acc + in[idx];    // conv + identity
}
// ---------------------------------------------------------------- norms
__global__ void k_inorm_stats(const float* x, float* stats){  // grid=16, block=256
    int c=blockIdx.x;
    const float* xp = x + (size_t)c*HW;
    __shared__ float ssum[256], ssq[256];
    float s=0.f,q=0.f;
    for (int i=threadIdx.x;i<HW;i+=256){ float v=xp[i]; s+=v; q+=v*v; }
    ssum[threadIdx.x]=s; ssq[threadIdx.x]=q; __syncthreads();
    for (int st=128; st>0; st>>=1){
        if (threadIdx.x<st){ ssum[threadIdx.x]+=ssum[threadIdx.x+st]; ssq[threadIdx.x]+=ssq[threadIdx.x+st]; }
        __syncthreads();
    }
    if (threadIdx.x==0){ float m=ssum[0]/(float)HW; stats[2*c]=m; stats[2*c+1]=ssq[0]/(float)HW - m*m; }
}
__global__ void k_inorm_apply(float* x, const float* stats, const float* g, const float* b){
    int i=blockIdx.x*256+threadIdx.x; if(i>=16*HW) return;
    int c=i/HW;
    x[i] = (x[i]-stats[2*c])*rsqrtf(stats[2*c+1]+1e-5f)*g[c]+b[c];
}
__global__ void k_layernorm(const float* x, const float* w, const float* b, float* out){
    int t=blockIdx.x*256+threadIdx.x; if(t>=LTOK) return;
    const float* xp=x+(size_t)t*CF;
    float mu=0.f;
    for (int c=0;c<CF;++c) mu+=xp[c];
    mu*= (1.f/CF);
    float var=0.f;
    for (int c=0;c<CF;++c){ float d=xp[c]-mu; var+=d*d; }
    var *= (1.f/CF);
    float inv=rsqrtf(var+1e-5f);
    float* op=out+(size_t)t*CF;
    for (int c=0;c<CF;++c) op[c]=(xp[c]-mu)*inv*w[c]+b[c];
}
// ---------------------------------------------------------------- mamba pieces
__global__ void k_causal_conv_silu(const float* __restrict__ x, int sx,
                                   const float* __restrict__ w, const float* __restrict__ b,
                                   float* __restrict__ out)
{
    int i=blockIdx.x*256+threadIdx.x; if(i>=LTOK*DI) return;
    int t=i/DI, d=i-t*DI;
    float acc=b[d];
#pragma unroll
    for (int k=0;k<4;++k){
        int tt=t+k-3;
        if (tt>=0) acc += x[(size_t)tt*sx+d]*w[d*4+k];
    }
    out[i]=siluf(acc);
}
__global__ void k_pad_weightT(const float* __restrict__ w, float* __restrict__ o){
    // w: (18,64) row-major -> o: (64,32) with cols>=18 zero
    int i=blockIdx.x*256+threadIdx.x; if(i>=DI*32) return;
    int k=i/32, n=i-k*32;
    o[i] = (n<18) ? w[n*DI+k] : 0.f;
}
__global__ void k_dtproj(const float* __restrict__ xdbl, const float* __restrict__ dtw,
                         const float* __restrict__ dtb, float* __restrict__ dt)
{
    int i=blockIdx.x*256+threadIdx.x; if(i>=LTOK*DI) return;
    int t=i/DI, d=i-t*DI;
    float v = xdbl[(size_t)t*32+0]*dtw[d*2+0] + xdbl[(size_t)t*32+1]*dtw[d*2+1] + dtb[d];
    dt[i] = (v > 20.f) ? v : log1pf(expf(v));
}
__global__ void k_scan(const float* __restrict__ xi, const float* __restrict__ dt,
                       const float* __restrict__ xdbl, const float* __restrict__ A_log,
                       const float* __restrict__ Dp, float* __restrict__ y)
{
    int d=threadIdx.x; if(d>=DI) return;
    float A[DSTATE], h[DSTATE];
#pragma unroll
    for (int n=0;n<DSTATE;++n){ A[n]=-expf(A_log[d*DSTATE+n]); h[n]=0.f; }
    const float Dd=Dp[d];
    for (int t=0;t<LTOK;++t){
        float dtv=dt[(size_t)t*DI+d];
        float xv =xi[(size_t)t*DI+d];
        float dx = dtv*xv;
        float acc=0.f;
        const float* row = xdbl + (size_t)t*32;
#pragma unroll
        for (int n=0;n<DSTATE;++n){
            h[n] = expf(dtv*A[n])*h[n] + dx*row[2+n];
            acc += h[n]*row[10+n];
        }
        y[(size_t)t*DI+d] = acc + xv*Dd;
    }
}
__global__ void k_gate(const float* __restrict__ y, const float* __restrict__ xz,
                       float* __restrict__ u)
{
    int i=blockIdx.x*256+threadIdx.x; if(i>=LTOK*DI) return;
    int t=i/DI, d=i-t*DI;
    u[i] = y[i]*siluf(xz[(size_t)t*128+64+d]);
}
// ---------------------------------------------------------------- CA head
__global__ void k_chmean(const float* y, float* s){ // grid=32, block=256
    int c=blockIdx.x;
    const float* yp = y + (size_t)c*HW;
    __shared__ float sm[256];
    float a=0.f;
    for (int i=threadIdx.x;i<HW;i+=256) a+=yp[i];
    sm[threadIdx.x]=a; __syncthreads();
    for (int st=128; st>0; st>>=1){ if(threadIdx.x<st) sm[threadIdx.x]+=sm[threadIdx.x+st]; __syncthreads(); }
    if (threadIdx.x==0) s[c]=sm[0]/(float)HW;
}
__global__ void k_ca_dense(const float* s, const float* d1w, const float* d1b,
                           const float* d2w, const float* d2b, float* sig)
{
    __shared__ float ss[32], hid[8];
    int tid=threadIdx.x;
    if (tid<32) ss[tid]=s[tid];
    __syncthreads();
    if (tid<8){
        float a=d1b[tid];
        for (int c=0;c<32;++c) a+=d1w[tid*32+c]*ss[c];
        hid[tid]=fmaxf(a,0.f);
    }
    __syncthreads();
    if (tid<32){
        float a=d2b[tid];
        for (int h=0;h<8;++h) a+=d2w[tid*8+h]*hid[h];
        sig[tid]=1.f/(1.f+expf(-a));
    }
}
__global__ void k_scale_add(const float* y, const float* sig, const float* x, float* o){
    int i=blockIdx.x*256+threadIdx.x; if(i>=CF*HW) return;
    int c=i/HW; o[i]=sig[c]*y[i]+x[i];
}

// ================================================================ host side
struct ConvP { const float *b, *w; };
struct LnP   { const float *b, *w; };
struct HinP  { ConvP c1, c2; const float *in_b, *in_g; };
struct EncP  { HinP blk[3]; ConvP c0; };
struct MamP  { const float *A_log,*D,*conv_b,*conv_extra_b,*conv_extra_w,*conv_w,
               *dt_b,*dt_w,*in_extra_w,*in_w,*out_w,*xproj_w; };
struct SingleP { LnP ln; MamP m; };
struct SwapP { LnP ln1, ln2; MamP ms_m, pan_m; };
struct CrossP { ConvP dw; LnP ln1, ln2; MamP m; };
struct CaP { ConvP d1,d2,p1,p2; };
struct RefineP { CaP ca1, ca2; ConvP cin, clast; };
struct Ws { float *XZ,*XI,*EX,*SRC,*XDBL,*DT,*Y,*U,*BPAD; };

static void gemm(const float* A,int sAm,int sAk,const float* B,int sBk,int sBn,
                 const float* bias,float* C,int sCm,int sCn,int M,int N,int K,hipStream_t s)
{
    if (N % 64 == 0) {
        dim3 g(M/16, N/64);
        k_wmma_gemm_nt<4><<<g, 32, 0, s>>>(A,sAm,sAk,B,sBk,sBn,bias,C,sCm,sCn,K);
    } else {                       // N % 32 == 0 for all call sites
        dim3 g(M/16, N/32);
        k_wmma_gemm_nt<2><<<g, 32, 0, s>>>(A,sAm,sAk,B,sBk,sBn,bias,C,sCm,sCn,K);
    }
}

static void run_mamba(const float* x, const float* extra, const MamP& m,
                      float* out, const Ws& w, hipStream_t s)
{
    // xz = x @ in_w.T  (L,32)x(32,128)
    gemm(x,32,1, m.in_w,1,32, nullptr, w.XZ,128,1, LTOK,128,32, s);
    k_causal_conv_silu<<<NBLK(LTOK*DI),256,0,s>>>(w.XZ,128, m.conv_w, m.conv_b, w.XI);
    const float* src = w.XI;
    if (extra){
        gemm(extra,32,1, m.in_extra_w,1,32, nullptr, w.EX,64,1, LTOK,64,32, s);
        k_causal_conv_silu<<<NBLK(LTOK*DI),256,0,s>>>(w.EX,64, m.conv_extra_w, m.conv_extra_b, w.SRC);
        src = w.SRC;
    }
    k_pad_weightT<<<NBLK(DI*32),256,0,s>>>(m.xproj_w, w.BPAD);
    gemm(src,64,1, w.BPAD,32,1, nullptr, w.XDBL,32,1, LTOK,32,64, s);
    k_dtproj<<<NBLK(LTOK*DI),256,0,s>>>(w.XDBL, m.dt_w, m.dt_b, w.DT);
    k_scan<<<1,64,0,s>>>(w.XI, w.DT, w.XDBL, m.A_log, m.D, w.Y);
    k_gate<<<NBLK(LTOK*DI),256,0,s>>>(w.Y, w.XZ, w.U);
    gemm(w.U,64,1, m.out_w,1,64, nullptr, out,32,1, LTOK,32,64, s);
}

extern "C" void kernel_launch(void* const* d_in, const int* in_sizes, int n_in,
                              void* d_out, int out_size, void* d_ws, size_t ws_size,
                              hipStream_t stream)
{
    (void)in_sizes; (void)n_in; (void)out_size; (void)ws_size;
    int ii = 0;
    auto nx   = [&]() -> const float* { return (const float*)d_in[ii++]; };
    auto rdCv = [&](){ ConvP c; c.b=nx(); c.w=nx(); return c; };
    auto rdLn = [&](){ LnP l; l.b=nx(); l.w=nx(); return l; };
    auto rdMam = [&](bool cross){
        MamP m{};
        m.A_log=nx(); m.D=nx(); m.conv_b=nx();
        if (cross){ m.conv_extra_b=nx(); m.conv_extra_w=nx(); }
        m.conv_w=nx(); m.dt_b=nx(); m.dt_w=nx();
        if (cross) m.in_extra_w=nx();
        m.in_w=nx(); m.out_w=nx(); m.xproj_w=nx();
        return m;
    };
    auto rdHin = [&](){ HinP h; h.c1=rdCv(); h.c2=rdCv(); h.in_b=nx(); h.in_g=nx(); return h; };
    auto rdEnc = [&](){ EncP e; for(int i=0;i<3;++i) e.blk[i]=rdHin(); e.c0=rdCv(); return e; };
    auto rdSingle = [&](){ SingleP s; s.ln=rdLn(); s.m=rdMam(false); return s; };
    auto rdSwap = [&](){ SwapP s; s.ln1=rdLn(); s.ln2=rdLn(); s.ms_m=rdMam(false); s.pan_m=rdMam(false); return s; };
    auto rdCross = [&](){ CrossP c; c.dw=rdCv(); c.ln1=rdLn(); c.ln2=rdLn(); c.m=rdMam(true); return c; };
    auto rdCa = [&](){ CaP c; c.d1=rdCv(); c.d2=rdCv(); c.p1=rdCv(); c.p2=rdCv(); return c; };

    // ---- inputs in jax tree_flatten (sorted-key) order
    const float* ms  = nx();
    const float* pan = nx();
    CrossP deep[5]; for (int i=0;i<5;++i) deep[i]=rdCross();
    EncP ms_enc = rdEnc();
    SingleP ms_fe[8]; for (int i=0;i<8;++i) ms_fe[i]=rdSingle();
    ConvP ms_tok = rdCv();
    EncP pan_enc = rdEnc();
    SingleP pan_fe[8]; for (int i=0;i<8;++i) pan_fe[i]=rdSingle();
    ConvP pan_tok = rdCv();
    RefineP ref; ref.ca1=rdCa(); ref.ca2=rdCa(); ref.cin=rdCv(); ref.clast=rdCv();
    ConvP sf1 = rdCv();
    ConvP sf2 = rdCv();
    SwapP swap1 = rdSwap();
    SwapP swap2 = rdSwap();

    // ---- workspace bump allocator
    char* base = (char*)d_ws; size_t off = 0;
    auto fal = [&](size_t n)->float*{ float* p=(float*)(base+off); off+=((n*4+255)/256)*256; return p; };
    float* MSBIC = fal(4*HW);
    float* IMG0 = fal(CF*HW); float* IMG1 = fal(CF*HW);
    float* IMG2 = fal(CF*HW); float* IMG3 = fal(CF*HW);
    float* CAT  = fal(2*CF*HW);
    float* MSF = fal(LTOK*CF); float* PANF = fal(LTOK*CF);
    float* RM  = fal(LTOK*CF); float* RP   = fal(LTOK*CF);
    float* T0  = fal(LTOK*CF); float* T1   = fal(LTOK*CF);
    float* P0  = fal(LTOK*CF); float* GTOK = fal(LTOK*CF);
    Ws w;
    w.XZ=fal(LTOK*128); w.XI=fal(LTOK*DI); w.EX=fal(LTOK*DI); w.SRC=fal(LTOK*DI);
    w.XDBL=fal(LTOK*32); w.DT=fal(LTOK*DI); w.Y=fal(LTOK*DI); w.U=fal(LTOK*DI);
    w.BPAD=fal(DI*32);
    float* STATS=fal(32); float* SV=fal(32); float* SIG=fal(32);
    hipStream_t s = stream;

    const int NC = CF*HW;
    // ---- ms_bic (nearest x4)
    k_upsample4<<<NBLK(4*HW),256,0,s>>>(ms, MSBIC);

    // ---- encoders + first tok
    auto run_encoder = [&](const float* inp, int Cin, const EncP& e)->const float*{
        float* A=IMG0; float* Bb=IMG1; float* Cc=IMG2;
        k_conv3x3<<<NBLK(NC),256,0,s>>>(inp, e.c0.w, e.c0.b, nullptr, A, Cin, CF, 0);
        for (int i=0;i<3;++i){
            const HinP& h=e.blk[i];
            k_conv3x3<<<NBLK(NC),256,0,s>>>(A, h.c1.w, h.c1.b, nullptr, Bb, CF, CF, 1);
            k_inorm_stats<<<16,256,0,s>>>(Bb, STATS);
            k_inorm_apply<<<NBLK(16*HW),256,0,s>>>(Bb, STATS, h.in_g, h.in_b);
            k_conv3x3<<<NBLK(NC),256,0,s>>>(Bb, h.c2.w, h.c2.b, A, Cc, CF, CF, 1);
            float* t=A; A=Cc; Cc=t;
        }
        return A;
    };
    auto run_tok = [&](const float* img, const ConvP& tk, float* out){
        gemm(img,1,HW, tk.w,1,CF, tk.b, out,CF,1, LTOK,CF,CF, s);
    };
    const float* msf_img = run_encoder(MSBIC, 4, ms_enc);
    run_tok(msf_img, ms_tok, MSF);
    const float* panf_img = run_encoder(pan, 1, pan_enc);
    run_tok(panf_img, pan_tok, PANF);

    // ---- single-block chains
    k_zero<<<NBLK(LTOK*CF),256,0,s>>>(RM, LTOK*CF);
    k_zero<<<NBLK(LTOK*CF),256,0,s>>>(RP, LTOK*CF);
    for (int i=0;i<8;++i){
        k_add<<<NBLK(LTOK*CF),256,0,s>>>(MSF, RM, RM, LTOK*CF);
        k_layernorm<<<NBLK(LTOK),256,0,s>>>(RM, ms_fe[i].ln.w, ms_fe[i].ln.b, T0);
        run_mamba(T0, nullptr, ms_fe[i].m, MSF, w, s);
    }
    for (int i=0;i<8;++i){
        k_add<<<NBLK(LTOK*CF),256,0,s>>>(PANF, RP, RP, LTOK*CF);
        k_layernorm<<<NBLK(LTOK),256,0,s>>>(RP, pan_fe[i].ln.w, pan_fe[i].ln.b, T0);
        run_mamba(T0, nullptr, pan_fe[i].m, PANF, w, s);
    }
    // ---- swap blocks
    auto run_swap = [&](const SwapP& sp){
        k_add<<<NBLK(LTOK*CF),256,0,s>>>(MSF, RM, RM, LTOK*CF);
        k_add<<<NBLK(LTOK*CF),256,0,s>>>(PANF, RP, RP, LTOK*CF);
        k_layernorm<<<NBLK(LTOK),256,0,s>>>(RM, sp.ln1.w, sp.ln1.b, T0);
        k_layernorm<<<NBLK(LTOK),256,0,s>>>(RP, sp.ln2.w, sp.ln2.b, T1);
        k_swap_mix<<<NBLK(LTOK*CF),256,0,s>>>(T0, T1, P0, GTOK);  // P0=ms_s, GTOK=pan_s
        run_mamba(P0,  nullptr, sp.ms_m,  MSF, w, s);
        run_mamba(GTOK,nullptr, sp.pan_m, PANF, w, s);
    };
    run_swap(swap1);
    run_swap(swap2);

    // ---- shallow fusion (sf1/sf2)
    k_tok_to_img<<<NBLK(NC),256,0,s>>>(MSF, IMG0);
    k_tok_to_img<<<NBLK(NC),256,0,s>>>(PANF, IMG1);
    hipMemcpyAsync(CAT,        IMG0, (size_t)NC*4, hipMemcpyDeviceToDevice, s);
    hipMemcpyAsync(CAT + NC,   IMG1, (size_t)NC*4, hipMemcpyDeviceToDevice, s);
    k_conv3x3<<<NBLK(NC),256,0,s>>>(CAT, sf1.w, sf1.b, IMG0, IMG2, 2*CF, CF, 0);
    hipMemcpyAsync(CAT,        IMG1, (size_t)NC*4, hipMemcpyDeviceToDevice, s);
    hipMemcpyAsync(CAT + NC,   IMG2, (size_t)NC*4, hipMemcpyDeviceToDevice, s);
    k_conv3x3<<<NBLK(NC),256,0,s>>>(CAT, sf2.w, sf2.b, IMG1, IMG3, 2*CF, CF, 0);
    run_tok(IMG2, ms_tok, MSF);
    run_tok(IMG3, pan_tok, PANF);

    // ---- deep cross-mamba blocks
    k_zero<<<NBLK(LTOK*CF),256,0,s>>>(RM, LTOK*CF);
    for (int i=0;i<5;++i){
        const CrossP& cp = deep[i];
        k_add<<<NBLK(LTOK*CF),256,0,s>>>(MSF, RM, RM, LTOK*CF);
        k_layernorm<<<NBLK(LTOK),256,0,s>>>(RM, cp.ln1.w, cp.ln1.b, T0);
        k_layernorm<<<NBLK(LTOK),256,0,s>>>(T0, cp.ln1.w, cp.ln1.b, T1);   // double LN
        k_layernorm<<<NBLK(LTOK),256,0,s>>>(PANF, cp.ln2.w, cp.ln2.b, T0);
        k_layernorm<<<NBLK(LTOK),256,0,s>>>(T0, cp.ln2.w, cp.ln2.b, P0);
        run_mamba(T1, P0, cp.m, GTOK, w, s);
        k_tok_to_img<<<NBLK(NC),256,0,s>>>(GTOK, IMG2);
        k_dwconv_add<<<NBLK(NC),256,0,s>>>(IMG2, cp.dw.w, cp.dw.b, IMG3);
        k_img_to_tok<<<NBLK(LTOK*CF),256,0,s>>>(IMG3, MSF);
    }

    // ---- refine head
    k_tok_to_img<<<NBLK(NC),256,0,s>>>(MSF, IMG0);
    k_conv3x3<<<NBLK(NC),256,0,s>>>(IMG0, ref.cin.w, ref.cin.b, nullptr, IMG1, CF, CF, 0);
    // ca1 on IMG1 -> IMG2
    k_conv3x3<<<NBLK(NC),256,0,s>>>(IMG1, ref.ca1.p1.w, ref.ca1.p1.b, nullptr, IMG2, CF, CF, 2);
    k_conv3x3<<<NBLK(NC),256,0,s>>>(IMG2, ref.ca1.p2.w, ref.ca1.p2.b, nullptr, IMG3, CF, CF, 0);
    k_chmean<<<32,256,0,s>>>(IMG3, SV);
    k_ca_dense<<<1,32,0,s>>>(SV, ref.ca1.d1.w, ref.ca1.d1.b, ref.ca1.d2.w, ref.ca1.d2.b, SIG);
    k_scale_add<<<NBLK(NC),256,0,s>>>(IMG3, SIG, IMG1, IMG2);
    // ca2 on IMG2 -> IMG1
    k_conv3x3<<<NBLK(NC),256,0,s>>>(IMG2, ref.ca2.p1.w, ref.ca2.p1.b, nullptr, IMG3, CF, CF, 2);
    k_conv3x3<<<NBLK(NC),256,0,s>>>(IMG3, ref.ca2.p2.w, ref.ca2.p2.b, nullptr, IMG0, CF, CF, 0);
    k_chmean<<<32,256,0,s>>>(IMG0, SV);
    k_ca_dense<<<1,32,0,s>>>(SV, ref.ca2.d1.w, ref.ca2.d1.b, ref.ca2.d2.w, ref.ca2.d2.b, SIG);
    k_scale_add<<<NBLK(NC),256,0,s>>>(IMG0, SIG, IMG2, IMG1);
    // clast 32->4 + ms_bic residual -> d_out
    k_conv3x3<<<NBLK(4*HW),256,0,s>>>(IMG1, ref.clast.w, ref.clast.b, MSBIC,
                                      (float*)d_out, CF, 4, 0);
}